// SpanMaskGenerator_13795434955369
// MI455X (gfx1250) — compile-verified
//
#include <hip/hip_runtime.h>

// SpanMaskGenerator for MI455X (gfx1250).
//
// Pure data-producer workload: 256 MiB of float mask output, ~32 KiB input,
// negligible compute -> HBM-store-bound, roofline ~11.5 us @ 23.3 TB/s.
// No contraction structure, so WMMA is inapplicable. Instead we use the
// CDNA5 Tensor Data Mover: each block computes a 2048-position chunk of
// context+target masks into LDS (16 KiB) and issues ONE tensor_store_from_lds
// describing a 2-row tile (row 0 -> context_mask, row 1 -> target_mask,
// tensor_dim0_stride = B*S elements), offloading all global stores to the DMA
// engine. Tail chunks are handled by TDM OOB semantics (tensor_dim0 = valid
// length; writes beyond the tensor are dropped).

#define NUM_SPANS 4
#define THREADS 256
#define POS_PER_THREAD 8
#define CHUNK (THREADS * POS_PER_THREAD)   // 2048 positions per block

typedef float v4f  __attribute__((ext_vector_type(4)));
typedef unsigned int u32x4 __attribute__((ext_vector_type(4)));
typedef int i32x4 __attribute__((ext_vector_type(4)));
typedef int i32x8 __attribute__((ext_vector_type(8)));

// Issue one TDM store: 2D tile, tile_dim0 = CHUNK elements (f32),
// tile_dim1 = 2 rows, rows separated in memory by stride_elems floats.
// LDS source: 2*CHUNK contiguous floats starting at lds_off.
static __device__ inline void tdm_store_2rows(unsigned lds_off,
                                              unsigned long long ga,     // global byte addr of row 0
                                              unsigned td0,              // valid elements in dim0
                                              unsigned long long stride_elems)
{
    // ---- D# group 0 (CDNA5 ISA) ----
    u32x4 g0;
    g0.x = 1u;                                              // count=1, user descriptor
    g0.y = lds_off;                                         // lds_addr (bytes)
    g0.z = (unsigned)ga;                                    // global_addr[31:0]
    g0.w = (unsigned)((ga >> 32) & 0x01FFFFFFull)           // global_addr[56:32]
         | (2u << 30);                                      // type = 2 ("image")

    // ---- D# group 1 (§8.4) ----
    i32x8 g1;
    g1[0] = (int)(2u << 16);                                // data_size=4B; no multicast/pad/iterate
    g1[1] = (int)(td0 << 16);                               // tensor_dim0[15:0] (atomic_barrier_addr=0)
    g1[2] = (int)((td0 >> 16) | (2u << 16));                // tensor_dim0[31:16] | tensor_dim1=2
    g1[3] = (int)((unsigned)CHUNK << 16);                   // tensor_dim1[31:16]=0 | tile_dim0=CHUNK
    g1[4] = 2;                                              // tile_dim1=2, tile_dim2=0 (2D tile)
    g1[5] = (int)(unsigned)(stride_elems & 0xFFFFFFFFull);  // tensor_dim0_stride[31:0]
    g1[6] = (int)(unsigned)((stride_elems >> 32) & 0xFFFFull); // stride[47:32] | dim1_stride[15:0]=0
    g1[7] = 0;                                              // dim1_stride[47:16]=0

    const i32x4 z4 = {0, 0, 0, 0};
#if defined(__clang_major__) && (__clang_major__ >= 23)
    // amdgpu-toolchain (clang-23 / therock-10.0 headers): 6-arg form
    const i32x8 z8 = {0, 0, 0, 0, 0, 0, 0, 0};
    __builtin_amdgcn_tensor_store_from_lds(g0, g1, z4, z4, z8, 0);
#else
    // ROCm 7.2 (clang-22): 5-arg form
    __builtin_amdgcn_tensor_store_from_lds(g0, g1, z4, z4, 0);
#endif
}

__global__ __launch_bounds__(THREADS) void span_mask_kernel(
    const float* __restrict__ rand_scales,
    const float* __restrict__ rand_starts,
    float* __restrict__ out,          // [2, B, S] flat: context then target
    int seq_len,
    unsigned long long target_base)   // (size_t)batch * seq_len (elements)
{
    // JAX/XLA computes the span parameters as separately-rounded f32 mul/add;
    // forbid FMA contraction so floor() sees bit-identical inputs.
    #pragma clang fp contract(off)

    __shared__ float smem[2 * CHUNK]; // [0,CHUNK): context ; [CHUNK,2*CHUNK): target

    // 2-D grid: no integer division needed.
    const int b     = (int)blockIdx.y;   // sample index
    const int chunk = (int)blockIdx.x;   // chunk within sample

    const float fS = (float)seq_len;

    // Span bounds: identical for every lane in this block -> the compiler
    // keeps the whole derivation on the scalar-float pipe (s_mul_f32/s_floor_f32).
    int istart[NUM_SPANS], iend[NUM_SPANS];
#pragma unroll
    for (int i = 0; i < NUM_SPANS; ++i) {
        const float rs = rand_scales[b * NUM_SPANS + i];
        const float rt = rand_starts[b * NUM_SPANS + i];
        const float scale = 0.15f + (rs * 0.1f);            // U(0.15,0.25), no FMA
        float span = floorf(scale * fS);
        span = fmaxf(span, 1.0f);
        const float maxs = fmaxf(fS - span, 0.0f);
        const float st   = floorf(rt * (maxs + 1.0f));
        const float en   = fminf(st + span, fS);
        istart[i] = __builtin_amdgcn_readfirstlane((int)st);
        iend[i]   = __builtin_amdgcn_readfirstlane((int)en);
    }

    const int l0   = (int)threadIdx.x * POS_PER_THREAD;  // offset within chunk
    const int pos0 = chunk * CHUNK + l0;                 // global position

    float t[POS_PER_THREAD];
#pragma unroll
    for (int j = 0; j < POS_PER_THREAD; ++j) {
        const int p = pos0 + j;
        bool hit = false;
#pragma unroll
        for (int i = 0; i < NUM_SPANS; ++i)
            hit = hit || ((p >= istart[i]) && (p < iend[i]));
        t[j] = hit ? 1.0f : 0.0f;
    }
    const v4f c0 = {1.0f - t[0], 1.0f - t[1], 1.0f - t[2], 1.0f - t[3]};
    const v4f c1 = {1.0f - t[4], 1.0f - t[5], 1.0f - t[6], 1.0f - t[7]};
    const v4f t0 = {t[0], t[1], t[2], t[3]};
    const v4f t1 = {t[4], t[5], t[6], t[7]};

    // Tile in LDS, row-major for the TDM: row 0 = context, row 1 = target.
    *(v4f*)&smem[l0]             = c0;   // ds_store_b128
    *(v4f*)&smem[l0 + 4]         = c1;
    *(v4f*)&smem[CHUNK + l0]     = t0;
    *(v4f*)&smem[CHUNK + l0 + 4] = t1;

    __syncthreads();   // all LDS writes visible before the DMA reads them

    // TDM ops issue per-wave and IGNORE EXEC, so the gate must be a *scalar*
    // branch that waves 1..7 jump over: compare an SGPR wave ID, not a lane mask.
    const unsigned wave_id =
        (unsigned)__builtin_amdgcn_readfirstlane((int)(threadIdx.x >> 5));
    if (wave_id == 0) {
        // Flat LDS pointer: low 32 bits are the LDS byte offset (ISA §10.2).
        const unsigned lds_off = (unsigned)(uintptr_t)&smem[0];
        const unsigned long long ga = (unsigned long long)(uintptr_t)
            (out + (unsigned long long)b * (unsigned long long)seq_len
                 + (unsigned long long)chunk * CHUNK);
        const int rem = seq_len - chunk * CHUNK;        // tail via TDM OOB drop
        const unsigned td0 = (unsigned)(rem < CHUNK ? rem : CHUNK);
        tdm_store_2rows(lds_off, ga, td0, target_base);
        // Keep this wave (and thus the workgroup's LDS) alive until DMA done.
        __builtin_amdgcn_s_wait_tensorcnt(0);
    }
}

extern "C" void kernel_launch(void* const* d_in, const int* in_sizes, int n_in,
                              void* d_out, int out_size, void* d_ws, size_t ws_size,
                              hipStream_t stream) {
    (void)n_in; (void)d_ws; (void)ws_size;
    const float* rand_scales = (const float*)d_in[0];
    const float* rand_starts = (const float*)d_in[1];
    float* out = (float*)d_out;

    // Scalars live in device memory; reconstruct on host from shapes:
    //   in_sizes[0] = batch * NUM_SPANS ; out_size = 2 * batch * seq_len.
    const int batch   = in_sizes[0] / NUM_SPANS;
    const int seq_len = (int)((size_t)out_size / (2u * (size_t)batch));
    const int chunks  = (seq_len + CHUNK - 1) / CHUNK;
    const unsigned long long target_base =
        (unsigned long long)batch * (unsigned long long)seq_len;

    dim3 grid((unsigned)chunks, (unsigned)batch);
    span_mask_kernel<<<grid, THREADS, 0, stream>>>(
        rand_scales, rand_starts, out, seq_len, target_base);
}